// PSRoIAlign_41832981463448
// MI455X (gfx1250) — compile-verified
//
#include <hip/hip_runtime.h>

// PSRoIAlign forward for MI455X (gfx1250, wave32).
//
// Problem constants from the reference:
//   OUT_SIZE = P = 7, GROUP_SIZE = GS = 7, OUTPUT_DIM = D = 10, SAMPLE_NUM = SN = 2
//   SPATIAL_SCALE = 0.0625, features (N, 490, 80, 80) f32, rois (R, 5) f32
//   output (R, D, P, P) f32
//
// Since GS == P, gh = ph and gw = pw, so channel c == d*49 + ph*7 + pw == tid.
// One block per ROI (uniform ROI -> scalar SMEM loads), 490 active threads,
// geometry deduplicated through LDS (ds_store_b128 / ds_load_b128 + split
// barrier), 16 independent global_load_b32 gathers per output to keep LOADcnt
// deep, coalesced global_store_b32 for the result.

#define PBINS 7
#define GSIZE 7
#define ODIM 10
#define SNUM 2
#define SCALE 0.0625f
#define FH 80
#define FW 80
#define FC (ODIM * GSIZE * GSIZE)        // 490
#define OUT_PER_ROI (ODIM * PBINS * PBINS) // 490
#define NENT (PBINS * SNUM)              // 14 entries per axis

struct alignas(16) GeomEntry {
    int   i0;   // low corner index (clamped)
    int   i1;   // high corner index (clamped)
    float l;    // fractional weight toward i1
    float m;    // validity multiplier: 1.0 or 0.0
};

__global__ __launch_bounds__(512)
void psroi_align_fwd(const float* __restrict__ feat,
                     const float* __restrict__ rois,
                     float* __restrict__ out)
{
    __shared__ GeomEntry sEY[NENT];  // y-axis: [ph*SN + s]
    __shared__ GeomEntry sEX[NENT];  // x-axis: [pw*SN + s]

    const int r = blockIdx.x;

    // Block-uniform ROI fetch -> scalar loads into SGPRs.
    const float* roi = rois + (size_t)r * 5;
    const float rb  = roi[0];
    const float rx1 = roi[1];
    const float ry1 = roi[2];
    const float rx2 = roi[3];
    const float ry2 = roi[4];

    const int   b   = (int)rb;
    const float x1  = rx1 * SCALE;
    const float y1  = ry1 * SCALE;
    const float x2  = (rx2 + 1.0f) * SCALE;
    const float y2  = (ry2 + 1.0f) * SCALE;
    const float rw  = fmaxf(x2 - x1, 0.1f);
    const float rh  = fmaxf(y2 - y1, 0.1f);
    const float bw  = rw * (1.0f / (float)PBINS);
    const float bh  = rh * (1.0f / (float)PBINS);

    const int tid = threadIdx.x;

    // Threads 0..13 build the 14 y-entries; threads 14..27 the 14 x-entries.
    if (tid < 2 * NENT) {
        const bool  isX   = (tid >= NENT);
        const int   i     = isX ? (tid - NENT) : tid;
        const int   p     = i / SNUM;
        const int   s     = i % SNUM;
        const float frac  = ((float)s + 0.5f) * (1.0f / (float)SNUM);
        const float start = isX ? x1 : y1;
        const float bsz   = isX ? bw : bh;
        const int   lim   = isX ? FW : FH;

        const float v  = start + bsz * ((float)p + frac);
        const float m  = (v > -1.0f && v < (float)lim) ? 1.0f : 0.0f;
        const float vc = fminf(fmaxf(v, 0.0f), (float)(lim - 1));
        const int   i0 = (int)floorf(vc);
        const int   i1 = (i0 + 1 < lim - 1) ? (i0 + 1) : (lim - 1);

        GeomEntry e;
        e.i0 = i0;
        e.i1 = i1;
        e.l  = vc - (float)i0;
        e.m  = m;
        if (isX) sEX[i] = e; else sEY[i] = e;   // ds_store_b128
    }

    __syncthreads();   // gfx1250 split barrier: s_barrier_signal / s_barrier_wait

    if (tid >= OUT_PER_ROI) return;   // lanes 490..511 idle

    const int pw = tid % PBINS;
    const int ph = (tid / PBINS) % PBINS;
    // channel c == tid (GS == P collapse)

    const float* fb = feat + ((size_t)b * FC + (size_t)tid) * (FH * FW);

    // Hoist the two x-entries (shared across both y samples): ds_load_b128 x2
    const GeomEntry ex0 = sEX[pw * SNUM + 0];
    const GeomEntry ex1 = sEX[pw * SNUM + 1];
    const GeomEntry ey0 = sEY[ph * SNUM + 0];
    const GeomEntry ey1 = sEY[ph * SNUM + 1];

    float acc = 0.0f;

    #pragma unroll
    for (int syi = 0; syi < SNUM; ++syi) {
        const GeomEntry ey = (syi == 0) ? ey0 : ey1;
        const float ly = ey.l;
        const float hy = 1.0f - ey.l;
        const float* row0 = fb + (size_t)ey.i0 * FW;
        const float* row1 = fb + (size_t)ey.i1 * FW;

        #pragma unroll
        for (int sxi = 0; sxi < SNUM; ++sxi) {
            const GeomEntry ex = (sxi == 0) ? ex0 : ex1;
            const float lx = ex.l;
            const float hx = 1.0f - ex.l;

            // 4 independent global_load_b32 gathers per sample; the fully
            // unrolled 2x2 loop lets the compiler issue all 16 loads before
            // the first s_wait_loadcnt.
            const float v00 = row0[ex.i0];
            const float v01 = row0[ex.i1];
            const float v10 = row1[ex.i0];
            const float v11 = row1[ex.i1];

            const float bil = hy * (hx * v00 + lx * v01)
                            + ly * (hx * v10 + lx * v11);
            acc += (ey.m * ex.m) * bil;
        }
    }

    // Coalesced store: consecutive tid -> consecutive addresses.
    out[(size_t)r * OUT_PER_ROI + tid] = acc * (1.0f / (float)(SNUM * SNUM));
}

extern "C" void kernel_launch(void* const* d_in, const int* in_sizes, int n_in,
                              void* d_out, int out_size, void* d_ws, size_t ws_size,
                              hipStream_t stream)
{
    (void)n_in; (void)out_size; (void)d_ws; (void)ws_size;

    const float* feat = (const float*)d_in[0];
    const float* rois = (const float*)d_in[1];
    float*       out  = (float*)d_out;

    const int R = in_sizes[1] / 5;   // rois is (R, 5)

    psroi_align_fwd<<<R, 512, 0, stream>>>(feat, rois, out);
}